// Regress_graph_gs_128849019554
// MI455X (gfx1250) — compile-verified
//
#include <hip/hip_runtime.h>
#include <math.h>

typedef __attribute__((ext_vector_type(2))) float v2f;
typedef __attribute__((ext_vector_type(8))) float v8f;

#define HID 64
#define NLAYERS 3

// ---------- degree / normalization ----------
__global__ void k_fill1(float* __restrict__ p, int n) {
  int i = blockIdx.x * blockDim.x + threadIdx.x;
  if (i < n) p[i] = 1.0f;                       // self-loop contributes 1 to deg
}

__global__ void k_deg(const long long* __restrict__ dst, float* __restrict__ deg, long long E) {
  long long e = (long long)blockIdx.x * blockDim.x + threadIdx.x;
  if (e < E) atomicAdd(&deg[dst[e]], 1.0f);
}

__global__ void k_rsqrt(float* __restrict__ p, int n) {
  int i = blockIdx.x * blockDim.x + threadIdx.x;
  if (i < n) p[i] = rsqrtf(p[i]);
}

// ---------- dense layer: O[n x 64] = H[n x 64] @ W[64 x 64], via V_WMMA_F32_16X16X4_F32 ----------
// One wave computes a 16-row x 64-col strip. blockDim.x = 256 (8 waves).
__global__ void k_gemm_wmma(const float* __restrict__ H, const float* __restrict__ W,
                            float* __restrict__ O, int n) {
  const int wave = threadIdx.x >> 5;
  const int lane = threadIdx.x & 31;
  const int tile = blockIdx.x * (blockDim.x >> 5) + wave;
  const int row0 = tile << 4;
  if (row0 >= n) return;                         // wave-uniform exit; EXEC stays full for WMMA
  const int half = lane >> 4;                    // 0: K={k,k+1}, 1: K={k+2,k+3}
  const int l16  = lane & 15;
  long long m = row0 + l16;
  if (m >= n) m = n - 1;                         // clamp (n is multiple of 16 here anyway)
  const float* __restrict__ hrow = H + m * HID;

  v8f acc0 = {}, acc1 = {}, acc2 = {}, acc3 = {};
#pragma unroll
  for (int k = 0; k < HID; k += 4) {
    const int kk = k + 2 * half;
    v2f a; a.x = hrow[kk]; a.y = hrow[kk + 1];
    const float* __restrict__ w0 = W + (long long)kk * HID;       // row K=kk
    const float* __restrict__ w1 = w0 + HID;                      // row K=kk+1
    v2f b0, b1, b2, b3;
    b0.x = w0[ 0 + l16]; b0.y = w1[ 0 + l16];
    b1.x = w0[16 + l16]; b1.y = w1[16 + l16];
    b2.x = w0[32 + l16]; b2.y = w1[32 + l16];
    b3.x = w0[48 + l16]; b3.y = w1[48 + l16];
    acc0 = __builtin_amdgcn_wmma_f32_16x16x4_f32(false, a, false, b0, (short)0, acc0, false, false);
    acc1 = __builtin_amdgcn_wmma_f32_16x16x4_f32(false, a, false, b1, (short)0, acc1, false, false);
    acc2 = __builtin_amdgcn_wmma_f32_16x16x4_f32(false, a, false, b2, (short)0, acc2, false, false);
    acc3 = __builtin_amdgcn_wmma_f32_16x16x4_f32(false, a, false, b3, (short)0, acc3, false, false);
  }
  // C/D layout: VGPR r, lanes 0-15 -> row M=r, lanes 16-31 -> row M=r+8, col N=l16
#pragma unroll
  for (int r = 0; r < 8; ++r) {
    long long row = row0 + r + 8 * half;
    if (row < n) {
      float* __restrict__ o = O + row * HID + l16;
      o[ 0] = acc0[r];
      o[16] = acc1[r];
      o[32] = acc2[r];
      o[48] = acc3[r];
    }
  }
}

// ---------- scatter init: self-loop + bias ----------
// o[i,f] = bias[f] + t[i,f] * dis[i]^2   (16 threads per node, float4 each)
__global__ void k_scatter_init(const float* __restrict__ t, const float* __restrict__ dis,
                               const float* __restrict__ bias, float* __restrict__ o, int n) {
  long long idx = (long long)blockIdx.x * blockDim.x + threadIdx.x;
  int i  = (int)(idx >> 4);
  int f4 = ((int)idx & 15) << 2;
  if (i >= n) return;
  float w = dis[i]; w *= w;
  float4 v = *(const float4*)(t + (long long)i * HID + f4);
  float4 r;
  r.x = fmaf(v.x, w, bias[f4 + 0]);
  r.y = fmaf(v.y, w, bias[f4 + 1]);
  r.z = fmaf(v.z, w, bias[f4 + 2]);
  r.w = fmaf(v.w, w, bias[f4 + 3]);
  *(float4*)(o + (long long)i * HID + f4) = r;
}

// ---------- edge scatter-add (L2-resident atomics) ----------
__global__ void k_scatter_edges(const float* __restrict__ t, const float* __restrict__ dis,
                                const long long* __restrict__ src, const long long* __restrict__ dst,
                                float* __restrict__ o, long long E) {
  long long idx = (long long)blockIdx.x * blockDim.x + threadIdx.x;
  long long e = idx >> 4;
  int f4 = ((int)idx & 15) << 2;
  if (e >= E) return;
  long long s = src[e], d = dst[e];
  float w = dis[s] * dis[d];
  float4 v = *(const float4*)(t + s * HID + f4);
  float* __restrict__ p = o + d * HID + f4;
  atomicAdd(p + 0, v.x * w);
  atomicAdd(p + 1, v.y * w);
  atomicAdd(p + 2, v.z * w);
  atomicAdd(p + 3, v.w * w);
}

// ---------- ELU ----------
__global__ void k_elu(float* __restrict__ h, long long n) {
  long long i = (long long)blockIdx.x * blockDim.x + threadIdx.x;
  if (i < n) {
    float v = h[i];
    h[i] = v > 0.0f ? v : expm1f(v);
  }
}

// ---------- pooling ----------
__global__ void k_zero(float* __restrict__ p, int n) {
  int i = blockIdx.x * blockDim.x + threadIdx.x;
  if (i < n) p[i] = 0.0f;
}

__global__ void k_pool(const float* __restrict__ h, const unsigned char* __restrict__ mask,
                       const long long* __restrict__ batch,
                       float* __restrict__ num, float* __restrict__ den, int n) {
  long long idx = (long long)blockIdx.x * blockDim.x + threadIdx.x;
  int i  = (int)(idx >> 4);
  int f4 = ((int)idx & 15) << 2;
  if (i >= n) return;
  if (!mask[i]) return;
  long long g = batch[i];
  float4 v = *(const float4*)(h + (long long)i * HID + f4);
  float* __restrict__ p = num + g * HID + f4;
  atomicAdd(p + 0, v.x);
  atomicAdd(p + 1, v.y);
  atomicAdd(p + 2, v.z);
  atomicAdd(p + 3, v.w);
  if (f4 == 0) atomicAdd(&den[g], 1.0f);
}

__global__ void k_final(const float* __restrict__ num, const float* __restrict__ den,
                        const float* __restrict__ ltw, const float* __restrict__ ltb,
                        float* __restrict__ out, int G) {
  int g = blockIdx.x * blockDim.x + threadIdx.x;
  if (g >= G) return;
  float d = fmaxf(den[g], 1.0f);
  float inv = 1.0f / d;
  float acc = 0.0f;
#pragma unroll 8
  for (int f = 0; f < HID; ++f) acc = fmaf(num[(long long)g * HID + f] * inv, ltw[f], acc);
  out[g] = acc + ltb[0];
}

extern "C" void kernel_launch(void* const* d_in, const int* in_sizes, int n_in,
                              void* d_out, int out_size, void* d_ws, size_t ws_size,
                              hipStream_t stream) {
  const float*          x     = (const float*)d_in[0];
  const long long*      ei    = (const long long*)d_in[1];
  const unsigned char*  mask  = (const unsigned char*)d_in[2];
  const long long*      batch = (const long long*)d_in[3];
  const float*          Ws    = (const float*)d_in[4];
  const float*          bs    = (const float*)d_in[5];
  const float*          ltw   = (const float*)d_in[6];
  const float*          ltb   = (const float*)d_in[7];
  float*                out   = (float*)d_out;

  const int       N = in_sizes[0] / HID;
  const long long E = in_sizes[1] / 2;
  const int       G = out_size;                 // output is [G, 1]
  const long long* src = ei;                    // edge_index[0]
  const long long* dst = ei + E;                // edge_index[1]

  // workspace layout (all f32)
  float* dis = (float*)d_ws;                    // N
  float* B0  = dis + N;                         // N*HID  (current h / scatter target)
  float* B1  = B0 + (size_t)N * HID;            // N*HID  (gemm output / messages)
  float* num = B1 + (size_t)N * HID;            // G*HID
  float* den = num + (size_t)G * HID;           // G   (contiguous with num)

  const int T = 256;
  const int nodeBlk  = (N + T - 1) / T;
  const int edgeBlk  = (int)((E + T - 1) / T);
  const int node16   = (int)(((long long)N * 16 + T - 1) / T);
  const int edge16   = (int)((E * 16 + T - 1) / T);
  const int eluBlk   = (int)(((long long)N * HID + T - 1) / T);
  const int tiles    = (N + 15) / 16;
  const int gemmBlk  = (tiles + 7) / 8;         // 8 waves per block, 1 tile per wave

  // GCN symmetric normalization (self-loops included via init to 1)
  k_fill1<<<nodeBlk, T, 0, stream>>>(dis, N);
  k_deg<<<edgeBlk, T, 0, stream>>>(dst, dis, E);
  k_rsqrt<<<nodeBlk, T, 0, stream>>>(dis, N);

  const float* hin = x;
  for (int l = 0; l < NLAYERS; ++l) {
    k_gemm_wmma<<<gemmBlk, T, 0, stream>>>(hin, Ws + (size_t)l * HID * HID, B1, N);
    k_scatter_init<<<node16, T, 0, stream>>>(B1, dis, bs + (size_t)l * HID, B0, N);
    k_scatter_edges<<<edge16, T, 0, stream>>>(B1, dis, src, dst, B0, E);
    k_elu<<<eluBlk, T, 0, stream>>>(B0, (long long)N * HID);
    hin = B0;
  }

  k_zero<<<(G * HID + G + T - 1) / T, T, 0, stream>>>(num, G * HID + G);
  k_pool<<<node16, T, 0, stream>>>(B0, mask, batch, num, den, N);
  k_final<<<(G + 63) / 64, 64, 0, stream>>>(num, den, ltw, ltb, out, G);
}